// EfficientAutoCorrelation_14456859919030
// MI455X (gfx1250) — compile-verified
//
#include <hip/hip_runtime.h>
#include <hip/hip_bf16.h>
#include <math.h>

// ---------------------------------------------------------------------------
// EfficientAutoCorrelation for MI455X (gfx1250).
// Core: four-step FFT (L = 16 x N2) where both DFT stages are complex GEMMs
// executed with V_WMMA_F32_16X16X4_F32 (4 real WMMAs per complex MAC chunk).
// DFT coefficients & twiddles generated in TRANS VALU (co-executes with WMMA).
// Global->LDS movement of the k sequence is done with the gfx1250 async
// engine (GLOBAL_LOAD_ASYNC_TO_LDS_B128, ASYNCcnt) overlapped with the q FFT.
// Workspace need: ~112 MiB (qp/kp/vp/corr + small stats region).
// ---------------------------------------------------------------------------

typedef float v8f __attribute__((ext_vector_type(8)));
typedef float v2f __attribute__((ext_vector_type(2)));
typedef int v4i_vs __attribute__((vector_size(4 * sizeof(int))));  // matches builtin sig

#define WMMA_F32_16x16x4(A, B, C) \
  __builtin_amdgcn_wmma_f32_16x16x4_f32(false, (A), false, (B), (short)0, (C), false, false)

#if __has_builtin(__builtin_amdgcn_global_load_async_to_lds_b128) && \
    __has_builtin(__builtin_amdgcn_s_wait_asynccnt)
#define HAVE_ASYNC_LDS 1
#define ASYNC_GLB(p) ((__attribute__((address_space(1))) v4i_vs*)(p))
#define ASYNC_LDS(p) ((__attribute__((address_space(3))) v4i_vs*)(p))
#else
#define HAVE_ASYNC_LDS 0
#endif

constexpr int Bc = 16;
constexpr int Lc = 1024;
constexpr int Hc = 8;
constexpr int Ec = 64;
constexpr float TWO_PI = 6.283185307179586476925f;

// ---------------------------------------------------------------------------
// In-place four-step transform of one length-LS complex sequence held in `buf`
// (scratch `scr`), executed by one wave32.  sgn = -1 forward, +1 inverse.
// Input/output both in natural order.
// ---------------------------------------------------------------------------
template <int LS>
__device__ inline void fft_pass(float2* __restrict__ buf, float2* __restrict__ scr,
                                int lane, float sgn, float oscale) {
  constexpr int N2 = LS / 16;
  const int nlo = lane & 15;       // column / M-row within a 16-tile
  const int khalf = lane >> 4;     // which K-half this lane supplies

  // ---- step 1: T[i2, j1] = sum_{j2} W_N2^{sgn*i2*j2} * x[j1 + 16*j2] -------
  for (int mt = 0; mt < N2 / 16; ++mt) {
    v8f accRe = {}; v8f accIm = {};
    const int m = 16 * mt + nlo;                       // output row i2
    for (int kc = 0; kc < N2 / 4; ++kc) {
      const int k0 = 4 * kc + 2 * khalf;               // j2 rows (this lane)
      float s0, c0, s1, c1;
      __sincosf(TWO_PI * (float)((m * k0) % N2) / (float)N2, &s0, &c0);
      __sincosf(TWO_PI * (float)((m * (k0 + 1)) % N2) / (float)N2, &s1, &c1);
      v2f aCos;  aCos.x = c0;        aCos.y = c1;
      v2f aPSin; aPSin.x = sgn * s0; aPSin.y = sgn * s1;   // +Ai
      v2f aMSin = -aPSin;                                  // -Ai
      const float2 e0 = buf[nlo + 16 * k0];
      const float2 e1 = buf[nlo + 16 * (k0 + 1)];
      v2f bRe; bRe.x = e0.x; bRe.y = e1.x;
      v2f bIm; bIm.x = e0.y; bIm.y = e1.y;
      accRe = WMMA_F32_16x16x4(aCos,  bRe, accRe);
      accRe = WMMA_F32_16x16x4(aMSin, bIm, accRe);
      accIm = WMMA_F32_16x16x4(aCos,  bIm, accIm);
      accIm = WMMA_F32_16x16x4(aPSin, bRe, accIm);
    }
    // twiddle by W_LS^{sgn*i2*j1}; store T at scr[j1*N2 + i2]
#pragma unroll
    for (int j = 0; j < 8; ++j) {
      const int i2 = 16 * mt + j + 8 * khalf;
      const int j1 = nlo;
      float sn, cs;
      __sincosf(TWO_PI * (float)((i2 * j1) % LS) / (float)LS, &sn, &cs);
      sn *= sgn;
      const float re = accRe[j], im = accIm[j];
      float2 t; t.x = re * cs - im * sn; t.y = re * sn + im * cs;
      scr[j1 * N2 + i2] = t;
    }
  }

  // ---- step 3: X[i1, i2] = sum_{j1} W_16^{sgn*i1*j1} * T[j1, i2] -----------
  for (int nt = 0; nt < N2 / 16; ++nt) {
    v8f accRe = {}; v8f accIm = {};
    const int col = 16 * nt + nlo;                     // i2 column
    for (int kc = 0; kc < 4; ++kc) {
      const int k0 = 4 * kc + 2 * khalf;               // j1 rows (this lane)
      float s0, c0, s1, c1;
      __sincosf(TWO_PI * (float)((nlo * k0) & 15) / 16.0f, &s0, &c0);
      __sincosf(TWO_PI * (float)((nlo * (k0 + 1)) & 15) / 16.0f, &s1, &c1);
      v2f aCos;  aCos.x = c0;        aCos.y = c1;
      v2f aPSin; aPSin.x = sgn * s0; aPSin.y = sgn * s1;
      v2f aMSin = -aPSin;
      const float2 e0 = scr[k0 * N2 + col];
      const float2 e1 = scr[(k0 + 1) * N2 + col];
      v2f bRe; bRe.x = e0.x; bRe.y = e1.x;
      v2f bIm; bIm.x = e0.y; bIm.y = e1.y;
      accRe = WMMA_F32_16x16x4(aCos,  bRe, accRe);
      accRe = WMMA_F32_16x16x4(aMSin, bIm, accRe);
      accIm = WMMA_F32_16x16x4(aCos,  bIm, accIm);
      accIm = WMMA_F32_16x16x4(aPSin, bRe, accIm);
    }
#pragma unroll
    for (int j = 0; j < 8; ++j) {
      const int i1 = j + 8 * khalf;
      const int i2 = 16 * nt + nlo;
      float2 o; o.x = accRe[j] * oscale; o.y = accIm[j] * oscale;
      buf[i2 + N2 * i1] = o;                           // natural order k = i2 + N2*i1
    }
  }
}

// ---------------------------------------------------------------------------
// One wave per (b,h,e) sequence: corr = irfft(rfft(q) * conj(rfft(k))).
// qp/kp are sequence-major [B,H,E,LS]; corr written [B,H,E,LS].
// ---------------------------------------------------------------------------
template <int LS>
__global__ __launch_bounds__(64)
void fft_corr_kernel(const float* __restrict__ qp, const float* __restrict__ kp,
                     float* __restrict__ corr) {
  constexpr int NW = 2;                                // waves (sequences) per block
  __shared__ float2 s_buf[NW][LS];
  __shared__ float2 s_scr[NW][LS];
#if HAVE_ASYNC_LDS
  __shared__ float s_stg[NW][LS];                      // contiguous async staging
#endif
  const int wave = threadIdx.x >> 5;
  const int lane = threadIdx.x & 31;
  const long seq = (long)blockIdx.x * NW + wave;
  float2* buf = s_buf[wave];
  float2* scr = s_scr[wave];
  const float* __restrict__ q = qp + seq * LS;
  const float* __restrict__ k = kp + seq * LS;

#if HAVE_ASYNC_LDS
  float* stg = s_stg[wave];
  constexpr int CHUNK = 128;                           // 32 lanes x 16 B
  // queue async copy q -> stage (ASYNCcnt)
#pragma unroll
  for (int i = 0; i < LS / CHUNK; ++i)
    __builtin_amdgcn_global_load_async_to_lds_b128(
        ASYNC_GLB(q + i * CHUNK + lane * 4),
        ASYNC_LDS(stg + i * CHUNK + lane * 4), 0, 0);
  __builtin_amdgcn_s_wait_asynccnt(0);
  // expand to complex
#pragma unroll
  for (int i = 0; i < LS / 32; ++i) {
    float2 z; z.x = stg[lane + 32 * i]; z.y = 0.0f; buf[lane + 32 * i] = z;
  }
  // fence DS reads of stage before the async engine overwrites it with k
  asm volatile("s_wait_dscnt 0x0" ::: "memory");
  // queue async copy k -> stage; it lands while the q FFT runs below
#pragma unroll
  for (int i = 0; i < LS / CHUNK; ++i)
    __builtin_amdgcn_global_load_async_to_lds_b128(
        ASYNC_GLB(k + i * CHUNK + lane * 4),
        ASYNC_LDS(stg + i * CHUNK + lane * 4), 0, 0);
#else
  __builtin_prefetch(k, 0, 1);                         // global_prefetch_b8: warm k
  __builtin_prefetch(k + LS / 2, 0, 1);
#pragma unroll
  for (int i = 0; i < LS / 32; ++i) {
    float2 z; z.x = q[lane + 32 * i]; z.y = 0.0f; buf[lane + 32 * i] = z;
  }
#endif

  // FFT(q)  (overlapped with async k copy when available)
  fft_pass<LS>(buf, scr, lane, -1.0f, 1.0f);
  float2 qh[LS / 32];
#pragma unroll
  for (int i = 0; i < LS / 32; ++i) qh[i] = buf[lane + 32 * i];

  // FFT(k)
#if HAVE_ASYNC_LDS
  __builtin_amdgcn_s_wait_asynccnt(0);
#pragma unroll
  for (int i = 0; i < LS / 32; ++i) {
    float2 z; z.x = stg[lane + 32 * i]; z.y = 0.0f; buf[lane + 32 * i] = z;
  }
#else
#pragma unroll
  for (int i = 0; i < LS / 32; ++i) {
    float2 z; z.x = k[lane + 32 * i]; z.y = 0.0f; buf[lane + 32 * i] = z;
  }
#endif
  fft_pass<LS>(buf, scr, lane, -1.0f, 1.0f);

  // spectrum: c = qh * conj(kh)
#pragma unroll
  for (int i = 0; i < LS / 32; ++i) {
    const float2 kh = buf[lane + 32 * i];
    float2 c;
    c.x = qh[i].x * kh.x + qh[i].y * kh.y;
    c.y = qh[i].y * kh.x - qh[i].x * kh.y;
    buf[lane + 32 * i] = c;
  }

  // inverse FFT (scaled), take real part
  fft_pass<LS>(buf, scr, lane, +1.0f, 1.0f / (float)LS);
  float* __restrict__ out = corr + seq * LS;
#pragma unroll
  for (int i = 0; i < LS / 32; ++i) out[lane + 32 * i] = buf[lane + 32 * i].x;
}

// ---------------------------------------------------------------------------
// Pool (mean over `scale`) + transpose [B,L,H,E] -> [B,H,E,Lp] (seq-major).
// ---------------------------------------------------------------------------
__global__ void pool_transpose_kernel(const float* __restrict__ in, float* __restrict__ outT,
                                      int Lp, int scale) {
  const long n = (long)blockIdx.x * blockDim.x + threadIdx.x;
  const long total = (long)Bc * Hc * Ec * Lp;
  if (n >= total) return;
  const int l = (int)(n % Lp); long t = n / Lp;
  const int e = (int)(t % Ec); t /= Ec;
  const int h = (int)(t % Hc);
  const int b = (int)(t / Hc);
  float s = 0.0f;
  for (int j = 0; j < scale; ++j)
    s += in[(((long)b * (Lp * scale) + (long)l * scale + j) * Hc + h) * Ec + e];
  outT[n] = s / (float)scale;
}

// Pool keeping [B,Lp,H,E] layout (for values).
__global__ void pool_kernel(const float* __restrict__ in, float* __restrict__ out,
                            int Lp, int scale) {
  const long n = (long)blockIdx.x * blockDim.x + threadIdx.x;
  const long total = (long)Bc * Lp * Hc * Ec;
  if (n >= total) return;
  const int e = (int)(n % Ec); long t = n / Ec;
  const int h = (int)(t % Hc); t /= Hc;
  const int l = (int)(t % Lp);
  const int b = (int)(t / Lp);
  float s = 0.0f;
  for (int j = 0; j < scale; ++j)
    s += in[(((long)b * (Lp * scale) + (long)l * scale + j) * Hc + h) * Ec + e];
  out[n] = s / (float)scale;
}

// ---------------------------------------------------------------------------
// Per-batch sum / sum-of-squares over corr[b] (for energy & unbiased var).
// ---------------------------------------------------------------------------
__global__ void stats_kernel(const float* __restrict__ corr, float* __restrict__ sum,
                             float* __restrict__ sumsq, int Ls) {
  __shared__ float s1[256];
  __shared__ float s2[256];
  const int b = blockIdx.x;
  const long N = (long)Hc * Ec * Ls;
  const float* p = corr + (long)b * N;
  float a = 0.0f, q = 0.0f;
  for (long i = threadIdx.x; i < N; i += blockDim.x) {
    const float x = p[i];
    a += x; q += x * x;
  }
  s1[threadIdx.x] = a; s2[threadIdx.x] = q;
  __syncthreads();
  for (int o = 128; o > 0; o >>= 1) {
    if ((int)threadIdx.x < o) { s1[threadIdx.x] += s1[threadIdx.x + o]; s2[threadIdx.x] += s2[threadIdx.x + o]; }
    __syncthreads();
  }
  if (threadIdx.x == 0) { sum[b] = s1[0]; sumsq[b] = s2[0]; }
}

// Channel mean cm[b,l] = mean_{h,e} corr[b,h,e,l]
__global__ void cm_kernel(const float* __restrict__ corr, float* __restrict__ cm, int Ls) {
  const long n = (long)blockIdx.x * blockDim.x + threadIdx.x;
  if (n >= (long)Bc * Ls) return;
  const int l = (int)(n % Ls);
  const int b = (int)(n / Ls);
  const float* p = corr + ((long)b * Hc * Ec) * Ls + l;
  float s = 0.0f;
  for (int he = 0; he < Hc * Ec; ++he) s += p[(long)he * Ls];
  cm[n] = s / (float)(Hc * Ec);
}

// Interior-peak count of cm[b,:]
__global__ void pk_kernel(const float* __restrict__ cm, float* __restrict__ pk, int Ls) {
  __shared__ float sc[256];
  const int b = blockIdx.x;
  const float* p = cm + (long)b * Ls;
  float cnt = 0.0f;
  for (int l = 1 + threadIdx.x; l < Ls - 1; l += blockDim.x)
    if (p[l] > p[l - 1] && p[l] > p[l + 1]) cnt += 1.0f;
  sc[threadIdx.x] = cnt;
  __syncthreads();
  for (int o = 128; o > 0; o >>= 1) {
    if ((int)threadIdx.x < o) sc[threadIdx.x] += sc[threadIdx.x + o];
    __syncthreads();
  }
  if (threadIdx.x == 0) pk[b] = sc[0];
}

// Tiny MLP 3->32->16->1, sigmoid, k = clip(int(ratio*Ls), 1, Ls)
__global__ void mlp_kernel(const float* __restrict__ sum, const float* __restrict__ sumsq,
                           const float* __restrict__ pk,
                           const float* __restrict__ w1, const float* __restrict__ b1,
                           const float* __restrict__ w2, const float* __restrict__ b2,
                           const float* __restrict__ w3, const float* __restrict__ b3,
                           int Ls, int* __restrict__ kvals) {
  const int b = threadIdx.x;
  if (b >= Bc) return;
  float energy = 0.0f;
  for (int j = 0; j < Bc; ++j) energy += sumsq[j];
  energy /= (float)Bc * (float)Ls;
  const long N = (long)Hc * Ec * Ls;
  const float mean = sum[b] / (float)N;
  const float var = (sumsq[b] - (float)N * mean * mean) / (float)(N - 1);
  float f[3] = {energy, var, pk[b]};
  float h1[32];
#pragma unroll
  for (int i = 0; i < 32; ++i) {
    float a = b1[i];
    for (int j = 0; j < 3; ++j) a += w1[i * 3 + j] * f[j];
    h1[i] = fmaxf(a, 0.0f);
  }
  float h2[16];
#pragma unroll
  for (int i = 0; i < 16; ++i) {
    float a = b2[i];
    for (int j = 0; j < 32; ++j) a += w2[i * 32 + j] * h1[j];
    h2[i] = fmaxf(a, 0.0f);
  }
  float a = b3[0];
#pragma unroll
  for (int j = 0; j < 16; ++j) a += w3[j] * h2[j];
  const float ratio = 1.0f / (1.0f + __expf(-a));
  int kv = (int)(ratio * (float)Ls);                   // FACTOR = 1, trunc like astype(int32)
  kv = kv < 1 ? 1 : (kv > Ls ? Ls : kv);
  kvals[b] = kv;
}

__device__ inline bool is_peak(const float* p, long i, int l, int Ls) {
  if (l < 1 || l >= Ls - 1) return false;
  const float m = p[i];
  return (m > p[i - 1]) && (m > p[i + 1]);
}

// Per-batch total peak count
__global__ void peak_count_kernel(const float* __restrict__ corr, int* __restrict__ count, int Ls) {
  __shared__ int sc[256];
  const int b = blockIdx.x;
  const long N = (long)Hc * Ec * Ls;
  const float* p = corr + (long)b * N;
  int c = 0;
  for (long i = threadIdx.x; i < N; i += blockDim.x)
    c += is_peak(p, i, (int)(i % Ls), Ls) ? 1 : 0;
  sc[threadIdx.x] = c;
  __syncthreads();
  for (int o = 128; o > 0; o >>= 1) {
    if ((int)threadIdx.x < o) sc[threadIdx.x] += sc[threadIdx.x + o];
    __syncthreads();
  }
  if (threadIdx.x == 0) count[b] = sc[0];
}

// ---- exact k-th largest peak value via 4-pass radix select (per batch) ----
__device__ inline unsigned f2key(float x) {
  const unsigned u = __float_as_uint(x);
  return (u & 0x80000000u) ? ~u : (u | 0x80000000u);
}

__global__ void radix_init_kernel(const int* __restrict__ kvals,
                                  unsigned* __restrict__ prefix, int* __restrict__ krem) {
  const int b = threadIdx.x;
  if (b < Bc) { prefix[b] = 0u; krem[b] = kvals[b]; }
}

__global__ void radix_pass_kernel(const float* __restrict__ corr,
                                  unsigned* __restrict__ prefix, int* __restrict__ krem,
                                  int Ls, int pass) {
  __shared__ unsigned hist[256];
  const int b = blockIdx.x;
  const int shift = 24 - 8 * pass;
  const unsigned pfx = prefix[b];
  for (int i = threadIdx.x; i < 256; i += blockDim.x) hist[i] = 0u;
  __syncthreads();
  const long N = (long)Hc * Ec * Ls;
  const float* p = corr + (long)b * N;
  for (long i = threadIdx.x; i < N; i += blockDim.x) {
    const int l = (int)(i % Ls);
    const float m = p[i];
    const unsigned key = is_peak(p, i, l, Ls) ? f2key(m) : 0u;  // non-peaks = minimum key
    if (pass == 0 || (key >> (shift + 8)) == (pfx >> (shift + 8)))
      atomicAdd(&hist[(key >> shift) & 255u], 1u);
  }
  __syncthreads();
  if (threadIdx.x == 0) {
    int kk = krem[b];
    unsigned cum = 0u;
    int bsel = 0;
    for (int v = 255; v >= 0; --v) {
      const unsigned h = hist[v];
      if (cum + h >= (unsigned)kk) { bsel = v; kk -= (int)cum; break; }
      cum += h;
    }
    prefix[b] = pfx | ((unsigned)bsel << shift);
    krem[b] = kk;
  }
}

// ---------------------------------------------------------------------------
// Masked softmax over lags + multiply v.  One wave per (b,h,e).
// corr [B,H,E,Ls]; v & opool [B,Ls,H,E].
// ---------------------------------------------------------------------------
__global__ void softmax_agg_kernel(const float* __restrict__ corr, const float* __restrict__ v,
                                   const int* __restrict__ count, const int* __restrict__ kvals,
                                   const unsigned* __restrict__ prefix,
                                   float* __restrict__ opool, int Ls) {
  const int wave = threadIdx.x >> 5;
  const int lane = threadIdx.x & 31;
  const long s = (long)blockIdx.x * (blockDim.x / 32) + wave;
  if (s >= (long)Bc * Hc * Ec) return;
  const int e = (int)(s % Ec); long t = s / Ec;
  const int h = (int)(t % Hc);
  const int b = (int)(t / Hc);
  const float* p = corr + s * Ls;
  const bool keep = count[b] > kvals[b];
  const unsigned pk_ = prefix[b];
  const float thresh = __uint_as_float((pk_ & 0x80000000u) ? (pk_ & 0x7fffffffu) : ~pk_);

  float mx = -1e30f;
  for (int l = lane; l < Ls; l += 32) {
    const float c = p[l];
    const bool peak = is_peak(p, l, l, Ls);
    const bool fin = keep ? (peak && c >= thresh) : peak;
    mx = fmaxf(mx, fin ? c : 0.0f);
  }
  for (int o = 16; o > 0; o >>= 1) mx = fmaxf(mx, __shfl_xor(mx, o, 32));

  float se = 0.0f;
  for (int l = lane; l < Ls; l += 32) {
    const float c = p[l];
    const bool peak = is_peak(p, l, l, Ls);
    const bool fin = keep ? (peak && c >= thresh) : peak;
    se += __expf((fin ? c : 0.0f) - mx);
  }
  for (int o = 16; o > 0; o >>= 1) se += __shfl_xor(se, o, 32);
  const float inv = 1.0f / se;

  for (int l = lane; l < Ls; l += 32) {
    const float c = p[l];
    const bool peak = is_peak(p, l, l, Ls);
    const bool fin = keep ? (peak && c >= thresh) : peak;
    const float w = __expf((fin ? c : 0.0f) - mx) * inv;
    const long oidx = (((long)b * Ls + l) * Hc + h) * Ec + e;
    opool[oidx] = w * v[oidx];
  }
}

// Linear interpolate opool [B,Ls,H,E] up to L, scale by sw[swi], accumulate.
__global__ void interp_add_kernel(const float* __restrict__ opool, const float* __restrict__ sw,
                                  int swi, float* __restrict__ out, int Ls) {
  const long n = (long)blockIdx.x * blockDim.x + threadIdx.x;
  if (n >= (long)Bc * Lc * Hc * Ec) return;
  const float wgt = sw[swi];
  float val;
  if (Ls == Lc) {
    val = opool[n];
  } else {
    const int e = (int)(n % Ec); long t = n / Ec;
    const int h = (int)(t % Hc); t /= Hc;
    const int l = (int)(t % Lc);
    const int b = (int)(t / Lc);
    float src = ((float)l + 0.5f) * ((float)Ls / (float)Lc) - 0.5f;
    src = fmaxf(src, 0.0f);
    const int x0 = (int)floorf(src);
    const int x1 = (x0 + 1 < Ls) ? x0 + 1 : Ls - 1;
    const float lam = src - (float)x0;
    const long i0 = (((long)b * Ls + x0) * Hc + h) * Ec + e;
    const long i1 = (((long)b * Ls + x1) * Hc + h) * Ec + e;
    val = opool[i0] * (1.0f - lam) + opool[i1] * lam;
  }
  out[n] += wgt * val;
}

// ---------------------------------------------------------------------------
extern "C" void kernel_launch(void* const* d_in, const int* in_sizes, int n_in,
                              void* d_out, int out_size, void* d_ws, size_t ws_size,
                              hipStream_t stream) {
  const float* queries = (const float*)d_in[0];
  const float* keys    = (const float*)d_in[1];
  const float* values  = (const float*)d_in[2];
  const float* sweights= (const float*)d_in[3];
  const float* w1 = (const float*)d_in[4];
  const float* b1 = (const float*)d_in[5];
  const float* w2 = (const float*)d_in[6];
  const float* b2 = (const float*)d_in[7];
  const float* w3 = (const float*)d_in[8];
  const float* b3 = (const float*)d_in[9];
  float* out = (float*)d_out;

  // workspace layout (floats); total ~112 MiB
  const size_t SEQ = (size_t)Bc * Hc * Ec;   // 8192
  float* ws   = (float*)d_ws;
  float* qp   = ws;                          // SEQ*1024
  float* kp   = qp + SEQ * 1024;             // SEQ*1024
  float* vp   = kp + SEQ * 1024;             // SEQ*512 (scales 2,4 only)
  float* corr = vp + SEQ * 512;              // SEQ*1024
  float* opool = qp;                         // alias: qp is dead after fft kernel
  float* smallf = corr + SEQ * 1024;
  float* sum   = smallf;                     // 16
  float* sumsq = smallf + 16;                // 16
  float* pkf   = smallf + 32;                // 16
  float* cm    = smallf + 64;                // Bc*1024
  int*      kvals  = (int*)(cm + (size_t)Bc * 1024);
  int*      count  = kvals + 16;
  int*      krem   = count + 16;
  unsigned* prefix = (unsigned*)(krem + 16);

  (void)hipMemsetAsync(d_out, 0, (size_t)out_size * sizeof(float), stream);

  const int scales[3] = {1, 2, 4};
  for (int i = 0; i < 3; ++i) {
    const int s = scales[i];
    const int Ls = Lc / s;
    const long totT = (long)Bc * Hc * Ec * Ls;

    pool_transpose_kernel<<<(unsigned)((totT + 255) / 256), 256, 0, stream>>>(queries, qp, Ls, s);
    pool_transpose_kernel<<<(unsigned)((totT + 255) / 256), 256, 0, stream>>>(keys, kp, Ls, s);
    const float* vptr = values;
    if (s > 1) {
      pool_kernel<<<(unsigned)((totT + 255) / 256), 256, 0, stream>>>(values, vp, Ls, s);
      vptr = vp;
    }

    const unsigned fftBlocks = (unsigned)(SEQ / 2);    // 2 waves (sequences) per block
    if (Ls == 1024)      fft_corr_kernel<1024><<<fftBlocks, 64, 0, stream>>>(qp, kp, corr);
    else if (Ls == 512)  fft_corr_kernel<512><<<fftBlocks, 64, 0, stream>>>(qp, kp, corr);
    else                 fft_corr_kernel<256><<<fftBlocks, 64, 0, stream>>>(qp, kp, corr);

    stats_kernel<<<Bc, 256, 0, stream>>>(corr, sum, sumsq, Ls);
    cm_kernel<<<(unsigned)(((long)Bc * Ls + 255) / 256), 256, 0, stream>>>(corr, cm, Ls);
    pk_kernel<<<Bc, 256, 0, stream>>>(cm, pkf, Ls);
    mlp_kernel<<<1, 32, 0, stream>>>(sum, sumsq, pkf, w1, b1, w2, b2, w3, b3, Ls, kvals);
    peak_count_kernel<<<Bc, 256, 0, stream>>>(corr, count, Ls);
    radix_init_kernel<<<1, 32, 0, stream>>>(kvals, prefix, krem);
    for (int p = 0; p < 4; ++p)
      radix_pass_kernel<<<Bc, 256, 0, stream>>>(corr, prefix, krem, Ls, p);

    softmax_agg_kernel<<<(unsigned)(SEQ / 8), 256, 0, stream>>>(corr, vptr, count, kvals,
                                                                prefix, opool, Ls);
    interp_add_kernel<<<(unsigned)(((long)Bc * Lc * Hc * Ec + 255) / 256), 256, 0, stream>>>(
        opool, sweights, i, out, Ls);
  }
}